// BalancedBCE_17523466567777
// MI455X (gfx1250) — compile-verified
//
#include <hip/hip_runtime.h>
#include <hip/hip_bf16.h>

// Balanced BCE loss, fp32, B=64, 512x512 per sample.
// Pass 1: streaming kernel -> per-(sample,chunk) partial triples in d_ws.
// Pass 2: tiny kernel applies per-sample balance weights, emits scalar loss.
// Cross-lane wave reduction uses V_WMMA_F32_16X16X4_F32 (B = ones).
// Softplus computed with native v_exp_f32/v_log_f32 (2 trans + 2 mul) to keep
// the kernel on the HBM side of the roofline (~11 VALU/elem vs ~100 for ocml log1pf).

typedef __attribute__((ext_vector_type(2))) float v2f;
typedef __attribute__((ext_vector_type(4))) float v4f;
typedef __attribute__((ext_vector_type(8))) float v8f;

#define HW 262144                 // 512*512 elements per sample
#define CHUNKS 32                 // blocks per sample
#define TPB 256                   // 8 waves (wave32)
#define ELEMS_PER_BLOCK (HW / CHUNKS)          // 8192
#define ITERS (ELEMS_PER_BLOCK / (TPB * 4))    // 8 float4 loads per thread

#define LOG2E 1.4426950408889634f
#define LN2   0.6931471805599453f

// softplus(-a) for a >= 0:  ln(1 + 2^(-a*log2e)) = ln2 * log2(1 + 2^(-a*log2e))
__device__ __forceinline__ float softplus_neg(float a) {
    float e = __builtin_amdgcn_exp2f(-LOG2E * a);   // v_exp_f32
    return LN2 * __builtin_amdgcn_logf(1.0f + e);   // v_log_f32
}

// Exact 32-lane sum via one f32 WMMA:
// A[m][0] = v_m (lanes 0..15, VGPR0), A[m][2] = v_{m+16} (lanes 16..31, VGPR0),
// A[*][1] = A[*][3] = 0.  B = all ones => D[m][n] = v_m + v_{m+16}.
// Per-lane sum of the 8 D VGPRs gives rows 0..7 total (lanes 0..15) and
// rows 8..15 total (lanes 16..31); lane0 + lane16 is the full wave sum.
__device__ __forceinline__ float wave_reduce_wmma(float v) {
    v2f a; a.x = v;    a.y = 0.0f;
    v2f b; b.x = 1.0f; b.y = 1.0f;
    v8f c = {};
    c = __builtin_amdgcn_wmma_f32_16x16x4_f32(
        /*neg_a=*/false, a, /*neg_b=*/false, b,
        /*c_mod=*/(short)0, c, /*reuse_a=*/false, /*reuse_b=*/false);
    float s = c[0] + c[1] + c[2] + c[3] + c[4] + c[5] + c[6] + c[7];
    return __shfl(s, 0, 32) + __shfl(s, 16, 32);
}

__global__ __launch_bounds__(TPB) void bce_partial_kernel(
    const float* __restrict__ x, const float* __restrict__ t,
    float* __restrict__ wsP, float* __restrict__ wsN, float* __restrict__ wsC) {
    const unsigned b     = blockIdx.y;
    const unsigned chunk = blockIdx.x;
    const size_t base = (size_t)b * HW + (size_t)chunk * ELEMS_PER_BLOCK;
    const v4f* x4 = (const v4f*)(x + base);
    const v4f* t4 = (const v4f*)(t + base);

    float accT = 0.0f;   // sum of bce over all elements
    float accP = 0.0f;   // sum of t * bce  (positives; t is exactly 0 or 1)
    float accC = 0.0f;   // sum of t        (positive count)
#pragma unroll
    for (int it = 0; it < ITERS; ++it) {
        const int idx = it * TPB + (int)threadIdx.x;
        v4f xv = __builtin_nontemporal_load(&x4[idx]);   // stream once, NT hint
        v4f tv = __builtin_nontemporal_load(&t4[idx]);
#pragma unroll
        for (int k = 0; k < 4; ++k) {
            float xi = xv[k], ti = tv[k];
            // stable BCE-with-logits: max(x,0) - x*t + log1p(exp(-|x|))
            float bce = fmaf(xi, -ti, fmaxf(xi, 0.0f)) + softplus_neg(fabsf(xi));
            accT += bce;
            accP  = fmaf(ti, bce, accP);
            accC += ti;
        }
    }
    float accN = accT - accP;    // sum of bce over negatives

    __shared__ float sP[TPB], sN[TPB], sC[TPB];
    sP[threadIdx.x] = accP; sN[threadIdx.x] = accN; sC[threadIdx.x] = accC;
    __syncthreads();

    if (threadIdx.x < 32) {                 // wave 0 only: EXEC all-ones for WMMA
        float p = 0.0f, n = 0.0f, cc = 0.0f;
#pragma unroll
        for (int j = 0; j < TPB / 32; ++j) {
            p  += sP[threadIdx.x + 32 * j];
            n  += sN[threadIdx.x + 32 * j];
            cc += sC[threadIdx.x + 32 * j];
        }
        p  = wave_reduce_wmma(p);
        n  = wave_reduce_wmma(n);
        cc = wave_reduce_wmma(cc);
        if (threadIdx.x == 0) {
            const unsigned id = b * CHUNKS + chunk;
            wsP[id] = p; wsN[id] = n; wsC[id] = cc;
        }
    }
}

__global__ __launch_bounds__(256) void bce_final_kernel(
    const float* __restrict__ wsP, const float* __restrict__ wsN,
    const float* __restrict__ wsC, float* __restrict__ out, int B) {
    const int b = (int)threadIdx.x;          // one thread per sample
    __shared__ float aP[256], aN[256], aC[256];
    if (b < B) {
        float sp = 0.0f, sn = 0.0f, sc = 0.0f;
        for (int j = 0; j < CHUNKS; ++j) {
            sp += wsP[b * CHUNKS + j];
            sn += wsN[b * CHUNKS + j];
            sc += wsC[b * CHUNKS + j];
        }
        const float Nf = (float)HW;
        float tmean = sc / Nf;               // per-sample positive fraction
        aP[b] = (1.0f - tmean) * sp;         // beta_pos * pos_sum
        aN[b] = tmean * sn;                  // beta_neg * neg_sum
        aC[b] = sc;                          // per-sample positive count
    }
    __syncthreads();
    if (b == 0) {
        float P = 0.0f, Q = 0.0f, C = 0.0f;
        for (int i = 0; i < B; ++i) { P += aP[i]; Q += aN[i]; C += aC[i]; }
        float posCnt = C;
        float negCnt = (float)B * (float)HW - C;
        out[0] = P / posCnt + Q / negCnt;
    }
}

extern "C" void kernel_launch(void* const* d_in, const int* in_sizes, int n_in,
                              void* d_out, int out_size, void* d_ws, size_t ws_size,
                              hipStream_t stream) {
    const float* x = (const float*)d_in[0];
    const float* t = (const float*)d_in[1];
    float* ws = (float*)d_ws;

    const int B = in_sizes[0] / HW;          // 64 for the reference shapes
    const int nPart = B * CHUNKS;            // 2048 partial triples

    dim3 grid(CHUNKS, B);
    bce_partial_kernel<<<grid, TPB, 0, stream>>>(
        x, t, ws, ws + nPart, ws + 2 * nPart);
    bce_final_kernel<<<1, 256, 0, stream>>>(
        ws, ws + nPart, ws + 2 * nPart, (float*)d_out, B);
}